// TDLayer_2396591751779
// MI455X (gfx1250) — compile-verified
//
#include <hip/hip_runtime.h>
#include <hip/hip_bf16.h>

#define FLT_BIG 3.402823466e38f

typedef float v2f __attribute__((ext_vector_type(2)));
typedef float v8f __attribute__((ext_vector_type(8)));

static constexpr int B_ = 4;
static constexpr int N_ = 8192;
static constexpr int S_ = 2048;
static constexpr int K_ = 16;
static constexpr int MTOT = B_ * S_ * K_;   // 131072 = 2^17

// ---------------------------------------------------------------------------
// Kernel 1: Farthest Point Sampling. One block per batch; 1024 threads hold
// 8 points each (coords + running min-dist) entirely in registers. Per
// iteration: emit current "far" sample coords, update distances, block argmax.
// ---------------------------------------------------------------------------
__global__ __launch_bounds__(1024) void fps_kernel(const float* __restrict__ xyz,
                                                   float* __restrict__ out_newxyz) {
    int b = blockIdx.x;
    int t = threadIdx.x;
    const float* xb = xyz + (size_t)b * 3 * N_;
    float px[8], py[8], pz[8], dist[8];
#pragma unroll
    for (int j = 0; j < 8; ++j) {
        int i = t + 1024 * j;
        px[j] = xb[i];
        py[j] = xb[N_ + i];
        pz[j] = xb[2 * N_ + i];
        dist[j] = 1e10f;
    }
    __shared__ float sc[3];
    __shared__ float sV[32];
    __shared__ int   sI[32];
    __shared__ int   sFar;

    float* o1 = out_newxyz + (size_t)b * 3 * S_;
    int far = 0;
    for (int s = 0; s < S_; ++s) {
        // owner of "far" publishes its coords (registers -> LDS)
        if (t == (far & 1023)) {
            int j = far >> 10;
            sc[0] = px[j]; sc[1] = py[j]; sc[2] = pz[j];
        }
        __syncthreads();
        float cx = sc[0], cy = sc[1], cz = sc[2];
        if (t == 0) { o1[s] = cx; o1[S_ + s] = cy; o1[2 * S_ + s] = cz; }

        float bv = -1.0f; int bi = 0;
#pragma unroll
        for (int j = 0; j < 8; ++j) {
            float dx = px[j] - cx, dy = py[j] - cy, dz = pz[j] - cz;
            float dd = dx * dx + dy * dy + dz * dz;
            float nd = fminf(dist[j], dd);
            dist[j] = nd;
            if (nd > bv) { bv = nd; bi = t + 1024 * j; }   // strict > keeps lowest idx
        }
        // wave32 argmax (max value, tie -> lowest index)
        for (int m = 1; m < 32; m <<= 1) {
            float ov = __shfl_xor(bv, m, 32);
            int   oi = __shfl_xor(bi, m, 32);
            if (ov > bv || (ov == bv && oi < bi)) { bv = ov; bi = oi; }
        }
        if ((t & 31) == 0) { sV[t >> 5] = bv; sI[t >> 5] = bi; }
        __syncthreads();
        if (t < 32) {
            bv = sV[t]; bi = sI[t];
            for (int m = 1; m < 32; m <<= 1) {
                float ov = __shfl_xor(bv, m, 32);
                int   oi = __shfl_xor(bi, m, 32);
                if (ov > bv || (ov == bv && oi < bi)) { bv = ov; bi = oi; }
            }
            if (t == 0) sFar = bi;
        }
        __syncthreads();
        far = sFar;
    }
}

// ---------------------------------------------------------------------------
// Kernel 2: kNN. One 256-thread block per query; 32 distances per thread kept
// in registers; 16 rounds of block argmin-with-exclusion (matches lax.top_k
// ordering incl. lowest-index tie break).
// ---------------------------------------------------------------------------
__global__ __launch_bounds__(256) void knn_kernel(const float* __restrict__ xyz,
                                                  const float* __restrict__ newxyz,
                                                  int* __restrict__ knn) {
    int q = blockIdx.x;          // b * S + s
    int b = q >> 11;
    int s = q & (S_ - 1);
    int t = threadIdx.x;
    const float* xb = xyz + (size_t)b * 3 * N_;
    float qx = newxyz[((size_t)b * 3 + 0) * S_ + s];
    float qy = newxyz[((size_t)b * 3 + 1) * S_ + s];
    float qz = newxyz[((size_t)b * 3 + 2) * S_ + s];
    float qn = qx * qx + qy * qy + qz * qz;

    float d[32];
#pragma unroll
    for (int j = 0; j < 32; ++j) {
        int i = t + 256 * j;
        float px = xb[i], py = xb[N_ + i], pz = xb[2 * N_ + i];
        d[j] = (qn - 2.0f * (qx * px + qy * py + qz * pz)) + (px * px + py * py + pz * pz);
    }

    __shared__ float sV[8];
    __shared__ int   sI[8];
    __shared__ int   sWin;

    for (int r = 0; r < K_; ++r) {
        float bv = FLT_BIG; int bi = 0x7fffffff;
#pragma unroll
        for (int j = 0; j < 32; ++j) {
            if (d[j] < bv) { bv = d[j]; bi = t + 256 * j; }
        }
        for (int m = 1; m < 32; m <<= 1) {
            float ov = __shfl_xor(bv, m, 32);
            int   oi = __shfl_xor(bi, m, 32);
            if (ov < bv || (ov == bv && oi < bi)) { bv = ov; bi = oi; }
        }
        if ((t & 31) == 0) { sV[t >> 5] = bv; sI[t >> 5] = bi; }
        __syncthreads();
        if (t < 32) {
            bv = (t < 8) ? sV[t] : FLT_BIG;
            bi = (t < 8) ? sI[t] : 0x7fffffff;
            for (int m = 1; m < 32; m <<= 1) {
                float ov = __shfl_xor(bv, m, 32);
                int   oi = __shfl_xor(bi, m, 32);
                if (ov < bv || (ov == bv && oi < bi)) { bv = ov; bi = oi; }
            }
            if (t == 0) sWin = bi;
        }
        __syncthreads();
        int win = sWin;
        if (t == 0) knn[(size_t)q * K_ + r] = win;
        if ((win & 255) == t) d[win >> 8] = FLT_BIG;   // exclude chosen
        __syncthreads();
    }
}

// ---------------------------------------------------------------------------
// Kernel 3: gather + concat. Builds Xg [68 x MTOT] (rows 0-2: centered xyz,
// rows 3-66: grouped features, row 67: zero pad) and writes grouped_xyz_norm.
// ---------------------------------------------------------------------------
__global__ __launch_bounds__(256) void gather_kernel(const float* __restrict__ xyz,
                                                     const float* __restrict__ points,
                                                     const float* __restrict__ newxyz,
                                                     const int* __restrict__ knn,
                                                     float* __restrict__ Xg,
                                                     float* __restrict__ out_gxyz) {
    int m = blockIdx.x * 256 + threadIdx.x;        // < MTOT
    int k = m & (K_ - 1);
    int s = (m >> 4) & (S_ - 1);
    int b = m >> 15;
    int idx = knn[m];
#pragma unroll
    for (int c = 0; c < 3; ++c) {
        float v = xyz[((size_t)b * 3 + c) * N_ + idx] - newxyz[((size_t)b * 3 + c) * S_ + s];
        Xg[(size_t)c * MTOT + m] = v;
        out_gxyz[(((size_t)b * 3 + c) * S_ + s) * K_ + k] = v;
    }
    for (int c = 0; c < 64; ++c) {
        Xg[(size_t)(3 + c) * MTOT + m] = points[((size_t)b * 64 + c) * N_ + idx];
    }
    Xg[(size_t)67 * MTOT + m] = 0.0f;
}

// ---------------------------------------------------------------------------
// WMMA GEMM: Y[c][m] = sum_k W[c][k] * X[k][m] + bias[c], with per-channel
// sum / sum-of-squares accumulation for training-mode BatchNorm.
// Block = 8 waves; each wave owns one 16-column M-tile and all COUT rows.
// f32 WMMA 16x16x4: A lane layout M=lane%16, Kslot = vgpr + 2*(lane/16);
// B/C/D: N=lane%16, row = vgpr + rows/2*(lane/16).
// MODE 0: linear [c][MTOT] store. MODE 1: store into [B][COUT][S*K] output.
// ---------------------------------------------------------------------------
template <int CIN, int CINPAD, int COUT, int MODE>
__global__ __launch_bounds__(256) void gemm_wmma_kernel(const float* __restrict__ X,
                                                        const float* __restrict__ W,
                                                        const float* __restrict__ bias,
                                                        float* __restrict__ Y,
                                                        float* __restrict__ sums,
                                                        float* __restrict__ sumsq) {
    constexpr int NT = COUT / 16;
    int lane = threadIdx.x & 31;
    int wave = threadIdx.x >> 5;
    int col0 = (blockIdx.x * 8 + wave) * 16;
    int n = lane & 15;
    int h = lane >> 4;

    v8f acc[NT];
#pragma unroll
    for (int r = 0; r < NT; ++r)
#pragma unroll
        for (int e = 0; e < 8; ++e) acc[r][e] = 0.0f;

    for (int k0 = 0; k0 < CINPAD; k0 += 4) {
        int kb0 = k0 + 2 * h;
        int kb1 = kb0 + 1;
        v2f bvec;
        bvec.x = X[(size_t)kb0 * MTOT + col0 + n];
        bvec.y = X[(size_t)kb1 * MTOT + col0 + n];
#pragma unroll
        for (int r = 0; r < NT; ++r) {
            int row = r * 16 + n;
            v2f avec;
            avec.x = (kb0 < CIN) ? W[(size_t)row * CIN + kb0] : 0.0f;
            avec.y = (kb1 < CIN) ? W[(size_t)row * CIN + kb1] : 0.0f;
            acc[r] = __builtin_amdgcn_wmma_f32_16x16x4_f32(
                false, avec, false, bvec, (short)0, acc[r], false, false);
        }
    }

#pragma unroll
    for (int r = 0; r < NT; ++r) {
#pragma unroll
        for (int e = 0; e < 8; ++e) {
            int c = r * 16 + e + 8 * h;
            float y = acc[r][e] + bias[c];
            // BN statistics: reduce across the 16 lanes sharing channel c
            float s1 = y, s2 = y * y;
#pragma unroll
            for (int m = 1; m < 16; m <<= 1) {
                s1 += __shfl_xor(s1, m, 32);
                s2 += __shfl_xor(s2, m, 32);
            }
            if (n == 0) {
                atomicAdd(&sums[c], s1);
                atomicAdd(&sumsq[c], s2);
            }
            if (MODE == 0) {
                Y[(size_t)c * MTOT + col0 + n] = y;
            } else {
                int mcol = col0 + n;
                int b = mcol >> 15;
                int rem = mcol & (S_ * K_ - 1);
                Y[(size_t)b * (COUT * S_ * K_) + (size_t)c * (S_ * K_) + rem] = y;
            }
        }
    }
}

// ---------------------------------------------------------------------------
// BN (training-mode) + ReLU, in place. MODE 0: [C][MTOT] with C=64.
// MODE 1: [B][128][S*K].
// ---------------------------------------------------------------------------
template <int MODE>
__global__ __launch_bounds__(256) void bn_relu_kernel(float* __restrict__ Y,
                                                      const float* __restrict__ sums,
                                                      const float* __restrict__ sumsq,
                                                      const float* __restrict__ gamma,
                                                      const float* __restrict__ beta) {
    size_t i = (size_t)blockIdx.x * 256 + threadIdx.x;
    int c = (MODE == 0) ? (int)(i >> 17) : (int)((i >> 15) & 127);
    const float inv = 1.0f / (float)MTOT;
    float mean = sums[c] * inv;
    float var = sumsq[c] * inv - mean * mean;
    float y = Y[i];
    float yn = (y - mean) * rsqrtf(var + 1e-5f) * gamma[c] + beta[c];
    Y[i] = fmaxf(yn, 0.0f);
}

// ---------------------------------------------------------------------------
// Max pool over K (innermost 16 elements).
// ---------------------------------------------------------------------------
__global__ __launch_bounds__(256) void maxpool_kernel(const float* __restrict__ np4,
                                                      float* __restrict__ pooled) {
    size_t i = (size_t)blockIdx.x * 256 + threadIdx.x;   // < B*128*S
    const float* p = np4 + i * K_;
    float m = p[0];
#pragma unroll
    for (int k = 1; k < K_; ++k) m = fmaxf(m, p[k]);
    pooled[i] = m;
}

// ---------------------------------------------------------------------------
extern "C" void kernel_launch(void* const* d_in, const int* in_sizes, int n_in,
                              void* d_out, int out_size, void* d_ws, size_t ws_size,
                              hipStream_t stream) {
    const float* xyz    = (const float*)d_in[0];   // [B,3,N]
    const float* points = (const float*)d_in[1];   // [B,64,N]
    const float* w1     = (const float*)d_in[2];   // [64,67]
    const float* b1     = (const float*)d_in[3];
    const float* g1     = (const float*)d_in[4];
    const float* beta1  = (const float*)d_in[5];
    const float* w2     = (const float*)d_in[6];   // [128,64]
    const float* b2     = (const float*)d_in[7];
    const float* g2     = (const float*)d_in[8];
    const float* beta2  = (const float*)d_in[9];

    // output regions (concatenated, return order)
    float* out = (float*)d_out;
    float* out_newxyz = out;                                   // [B,3,S]      24576
    float* out_pooled = out + 24576;                           // [B,128,S]    1048576
    float* out_gxyz   = out + 24576 + 1048576;                 // [B,3,S,K]    393216
    float* out_np     = out + 24576 + 1048576 + 393216;        // [B,128,S,K]  16777216

    // workspace layout
    char* ws = (char*)d_ws;
    int*   knn  = (int*)ws;                                    // MTOT ints
    float* Xg   = (float*)(ws + (size_t)MTOT * 4);             // 68 x MTOT
    float* Y1   = (float*)(ws + (size_t)MTOT * 4 + (size_t)68 * MTOT * 4);  // 64 x MTOT
    float* stat = (float*)(ws + (size_t)MTOT * 4 + (size_t)(68 + 64) * MTOT * 4);
    float* sums1 = stat, *sumsq1 = stat + 64, *sums2 = stat + 128, *sumsq2 = stat + 256;

    hipMemsetAsync(stat, 0, 384 * sizeof(float), stream);

    fps_kernel<<<B_, 1024, 0, stream>>>(xyz, out_newxyz);
    knn_kernel<<<B_ * S_, 256, 0, stream>>>(xyz, out_newxyz, knn);
    gather_kernel<<<MTOT / 256, 256, 0, stream>>>(xyz, points, out_newxyz, knn, Xg, out_gxyz);

    gemm_wmma_kernel<67, 68, 64, 0><<<MTOT / 128, 256, 0, stream>>>(Xg, w1, b1, Y1, sums1, sumsq1);
    bn_relu_kernel<0><<<(64 * MTOT) / 256, 256, 0, stream>>>(Y1, sums1, sumsq1, g1, beta1);

    gemm_wmma_kernel<64, 64, 128, 1><<<MTOT / 128, 256, 0, stream>>>(Y1, w2, b2, out_np, sums2, sumsq2);
    bn_relu_kernel<1><<<(128 * MTOT) / 256, 256, 0, stream>>>(out_np, sums2, sumsq2, g2, beta2);

    maxpool_kernel<<<(B_ * 128 * S_) / 256, 256, 0, stream>>>(out_np, out_pooled);
}